// MoE_4123168604696
// MI455X (gfx1250) — compile-verified
//
#include <hip/hip_runtime.h>

#define D     1024
#define NEXP  8
#define TM    32                 // tokens per tile
#define KC    8                  // K rows per async chunk
#define WSTRIDE 144              // padded floats per staged W row (bank-conflict-free)
#define HT_FLOATS (D * (TM + 1))            // 1024*33 = 33792
#define SLICE_FLOATS (2 * KC * WSTRIDE)     // per-wave double buffer = 2304
#define LDS_FLOATS (HT_FLOATS + 8 * SLICE_FLOATS)   // 33792 + 18432 = 52224

typedef float v2f __attribute__((ext_vector_type(2)));
typedef float v8f __attribute__((ext_vector_type(8)));

// ------------------------------------------------ CDNA5 async helpers
__device__ __forceinline__ void async_w_chunk(const float* We, int kbase, int wv,
                                              int lane, unsigned dst_bytes) {
  // stage 8 rows x 128 cols of W into this wave's LDS slice: 8x b128 async DMA
  const float* g = We + (size_t)kbase * D + wv * 128 + lane * 4;
#pragma unroll
  for (int j = 0; j < KC; ++j) {
    asm volatile("global_load_async_to_lds_b128 %0, %1, off"
                 :: "v"(dst_bytes + (unsigned)(4 * WSTRIDE) * (unsigned)j),
                    "v"((unsigned long long)(uintptr_t)(g + (size_t)j * D))
                 : "memory");
  }
}
__device__ __forceinline__ void wait_async_le8() {
  asm volatile("s_wait_asynccnt 0x8" ::: "memory");
}
__device__ __forceinline__ void wait_async_0() {
  asm volatile("s_wait_asynccnt 0x0" ::: "memory");
}

// ---------------------------------------------------------------- init
__global__ void init_counts(int* __restrict__ counts_cursor) {
  int i = threadIdx.x;
  if (i < 4 * NEXP) counts_cursor[i] = 0;   // counts[2][8] + cursor[2][8]
}

// ---------------------------------------------------------------- gating
__global__ __launch_bounds__(256)
void gate_kernel(const float* __restrict__ x, const float* __restrict__ gW,
                 const float* __restrict__ gb, int* __restrict__ topi,
                 float* __restrict__ topw, int* __restrict__ counts, int T)
{
  int lane = threadIdx.x & 31;
  int wv   = threadIdx.x >> 5;
  int t = blockIdx.x * 8 + wv;
  if (t >= T) return;

  const float* xr = x + (size_t)t * D;
  float p[NEXP];
#pragma unroll
  for (int e = 0; e < NEXP; ++e) p[e] = 0.f;

  for (int d = lane; d < D; d += 32) {
    float xv = xr[d];
    const float* wr = gW + d * NEXP;
#pragma unroll
    for (int e = 0; e < NEXP; ++e) p[e] = fmaf(xv, wr[e], p[e]);
  }
#pragma unroll
  for (int e = 0; e < NEXP; ++e) {
#pragma unroll
    for (int off = 16; off > 0; off >>= 1)
      p[e] += __shfl_xor(p[e], off, 32);
  }

  if (lane == 0) {
    float lg[NEXP], pr[NEXP];
    float mx = -1e30f;
#pragma unroll
    for (int e = 0; e < NEXP; ++e) { lg[e] = p[e] + gb[e]; mx = fmaxf(mx, lg[e]); }
    float s = 0.f;
#pragma unroll
    for (int e = 0; e < NEXP; ++e) { pr[e] = expf(lg[e] - mx); s += pr[e]; }
    float inv = 1.f / s;
#pragma unroll
    for (int e = 0; e < NEXP; ++e) pr[e] *= inv;

    int i0 = 0;
#pragma unroll
    for (int e = 1; e < NEXP; ++e) if (pr[e] > pr[i0]) i0 = e;
    int i1 = -1; float bv = -1.f;
#pragma unroll
    for (int e = 0; e < NEXP; ++e)
      if (e != i0 && pr[e] > bv) { bv = pr[e]; i1 = e; }

    float a = pr[i0], b = pr[i1];
    float eb = expf(b - a);
    float w0 = 1.f / (1.f + eb);
    float w1 = eb * w0;

    topi[2 * t] = i0;  topi[2 * t + 1] = i1;
    topw[2 * t] = w0;  topw[2 * t + 1] = w1;
    atomicAdd(&counts[i0], 1);
    atomicAdd(&counts[NEXP + i1], 1);
  }
}

// ---------------------------------------------------------------- scan + tile build
__global__ void scan_kernel(const int* __restrict__ counts, int* __restrict__ offsets,
                            int* __restrict__ tileCount, int* __restrict__ tile_expert,
                            int* __restrict__ tile_start, int* __restrict__ tile_nvalid,
                            int maxTiles)
{
  if (threadIdx.x != 0 || blockIdx.x != 0) return;
  for (int s = 0; s < 2; ++s) {
    int off = 0, nt = 0;
    for (int e = 0; e < NEXP; ++e) {
      offsets[s * NEXP + e] = off;
      int c = counts[s * NEXP + e];
      for (int b = 0; b < c; b += TM) {
        tile_expert[s * maxTiles + nt] = e;
        tile_start [s * maxTiles + nt] = off + b;
        tile_nvalid[s * maxTiles + nt] = (c - b < TM) ? (c - b) : TM;
        ++nt;
      }
      off += c;
    }
    tileCount[s] = nt;
  }
}

// ---------------------------------------------------------------- scatter tokens
__global__ __launch_bounds__(256)
void scatter_kernel(const int* __restrict__ topi, const int* __restrict__ offsets,
                    int* __restrict__ cursor, int* __restrict__ lists, int T)
{
  int t = blockIdx.x * 256 + threadIdx.x;
  if (t >= T) return;
#pragma unroll
  for (int s = 0; s < 2; ++s) {
    int e = topi[2 * t + s];
    int pos = atomicAdd(&cursor[s * NEXP + e], 1);
    lists[(size_t)s * T + offsets[s * NEXP + e] + pos] = t;
  }
}

// ---------------------------------------------------------------- routed expert MLP
// block = 8 waves; tile = 32 tokens of one expert; wave wv owns N columns [wv*128, wv*128+128)
// W operand is DMA-staged per-wave into LDS (double-buffered async b128, ASYNCcnt-paced).
// pass1: h = relu(x @ W1 + b1) -> LDS transposed [n][m] stride 33
// pass2: y = h @ W2 + b2; out = (slot==0 ? store : add) of w * y
__global__ __launch_bounds__(256)
void moe_gemm_kernel(const float* __restrict__ x,
                     const float* __restrict__ W1, const float* __restrict__ b1,
                     const float* __restrict__ W2, const float* __restrict__ b2,
                     float* __restrict__ out,
                     const int* __restrict__ tileCount,
                     const int* __restrict__ tile_expert,
                     const int* __restrict__ tile_start,
                     const int* __restrict__ tile_nvalid,
                     const int* __restrict__ lists,
                     const float* __restrict__ topw,
                     int T, int maxTiles, int slot)
{
  extern __shared__ float lds[];           // [0,33792): hT ; [33792,52224): W slices
  __shared__ int   tokS[TM];
  __shared__ float wS[TM];

  int bid = blockIdx.x;
  if (bid >= tileCount[slot]) return;      // wave-uniform exit
  int e      = tile_expert[slot * maxTiles + bid];
  int lstart = tile_start [slot * maxTiles + bid];
  int nvalid = tile_nvalid[slot * maxTiles + bid];

  int tid = threadIdx.x, lane = tid & 31, wv = tid >> 5;
  if (tid < TM) {
    int idx = (tid < nvalid) ? tid : 0;
    int tok = lists[(size_t)slot * T + lstart + idx];
    tokS[tid] = tok;
    wS[tid]   = (tid < nvalid) ? topw[2 * tok + slot] : 0.f;
  }
  __syncthreads();

  int m     = lane & 15;
  int khalf = (lane >> 4) << 1;            // 0 or 2
  int ncol  = wv * 128 + m;
  int mbase = (lane >> 4) << 3;            // 0 or 8

  const int   SL   = HT_FLOATS + wv * SLICE_FLOATS;          // wave slice base (floats)
  const unsigned lds0 = __builtin_amdgcn_groupstaticsize();  // dynamic LDS starts here
  const unsigned slB  = lds0 + 4u * (unsigned)SL + (unsigned)(lane * 16);

  const float* W1e = W1 + (size_t)e * D * D;
  const float* W2e = W2 + (size_t)e * D * D;
  const float* ax0 = x + (size_t)tokS[m] * D + khalf;
  const float* ax1 = x + (size_t)tokS[m + 16] * D + khalf;

  v8f acc0[8] = {};
  v8f acc1[8] = {};

  // ---------------- pass 1: h = relu(x @ W1 + b1) ----------------
  async_w_chunk(W1e, 0, wv, lane, slB);                       // prefetch chunk 0
  for (int c = 0; c < D / KC; ++c) {
    int buf = c & 1;
    if (c + 1 < D / KC) {
      async_w_chunk(W1e, (c + 1) * KC, wv, lane, slB + 4u * KC * WSTRIDE * ((c + 1) & 1));
      wait_async_le8();                                       // chunk c landed
    } else {
      wait_async_0();
    }
    const float* wb = lds + SL + buf * (KC * WSTRIDE) + khalf * WSTRIDE + m;
#pragma unroll
    for (int kl = 0; kl < KC; kl += 4) {
      int kg = c * KC + kl;
      v2f a0 = *(const v2f*)(ax0 + kg);
      v2f a1 = *(const v2f*)(ax1 + kg);
#pragma unroll
      for (int t = 0; t < 8; ++t) {
        v2f bvv;
        bvv.x = wb[kl * WSTRIDE + t * 16];
        bvv.y = wb[kl * WSTRIDE + WSTRIDE + t * 16];
        acc0[t] = __builtin_amdgcn_wmma_f32_16x16x4_f32(
                    false, a0, false, bvv, (short)0, acc0[t], false, false);
        acc1[t] = __builtin_amdgcn_wmma_f32_16x16x4_f32(
                    false, a1, false, bvv, (short)0, acc1[t], false, false);
      }
    }
  }

  // prefetch pass-2 chunk 0 while we drain accumulators to LDS
  async_w_chunk(W2e, 0, wv, lane, slB);

  // bias + relu -> hT[n][m], stride 33 (conflict-free)
#pragma unroll
  for (int t = 0; t < 8; ++t) {
    int n = ncol + t * 16;
    float bias = b1[(size_t)e * D + n];
#pragma unroll
    for (int r = 0; r < 8; ++r) {
      float h0 = acc0[t][r] + bias;
      float h1 = acc1[t][r] + bias;
      lds[n * (TM + 1) + mbase + r]      = h0 > 0.f ? h0 : 0.f;
      lds[n * (TM + 1) + mbase + r + 16] = h1 > 0.f ? h1 : 0.f;
    }
  }
  __syncthreads();

  // ---------------- pass 2: y = h @ W2 + b2 ----------------
#pragma unroll
  for (int t = 0; t < 8; ++t) { acc0[t] = (v8f){}; acc1[t] = (v8f){}; }

  for (int c = 0; c < D / KC; ++c) {
    int buf = c & 1;
    if (c + 1 < D / KC) {
      async_w_chunk(W2e, (c + 1) * KC, wv, lane, slB + 4u * KC * WSTRIDE * ((c + 1) & 1));
      wait_async_le8();
    } else {
      wait_async_0();
    }
    const float* wb = lds + SL + buf * (KC * WSTRIDE) + khalf * WSTRIDE + m;
#pragma unroll
    for (int kl = 0; kl < KC; kl += 4) {
      int kk = c * KC + kl + khalf;
      v2f a0, a1;
      a0.x = lds[kk * (TM + 1) + m];
      a0.y = lds[(kk + 1) * (TM + 1) + m];
      a1.x = lds[kk * (TM + 1) + m + 16];
      a1.y = lds[(kk + 1) * (TM + 1) + m + 16];
#pragma unroll
      for (int t = 0; t < 8; ++t) {
        v2f bvv;
        bvv.x = wb[kl * WSTRIDE + t * 16];
        bvv.y = wb[kl * WSTRIDE + WSTRIDE + t * 16];
        acc0[t] = __builtin_amdgcn_wmma_f32_16x16x4_f32(
                    false, a0, false, bvv, (short)0, acc0[t], false, false);
        acc1[t] = __builtin_amdgcn_wmma_f32_16x16x4_f32(
                    false, a1, false, bvv, (short)0, acc1[t], false, false);
      }
    }
  }

  // ---------------- weighted scatter of y ----------------
#pragma unroll
  for (int t = 0; t < 8; ++t) {
    int n = ncol + t * 16;
    float bias = b2[(size_t)e * D + n];
#pragma unroll
    for (int r = 0; r < 8; ++r) {
      int mm = mbase + r;
      if (mm < nvalid) {
        float yv = (acc0[t][r] + bias) * wS[mm];
        size_t oi = (size_t)tokS[mm] * D + n;
        if (slot == 0) out[oi] = yv;
        else           out[oi] += yv;
      }
      int mm2 = mm + 16;
      if (mm2 < nvalid) {
        float yv = (acc1[t][r] + bias) * wS[mm2];
        size_t oi = (size_t)tokS[mm2] * D + n;
        if (slot == 0) out[oi] = yv;
        else           out[oi] += yv;
      }
    }
  }
}

// ---------------------------------------------------------------- launch
extern "C" void kernel_launch(void* const* d_in, const int* in_sizes, int n_in,
                              void* d_out, int out_size, void* d_ws, size_t ws_size,
                              hipStream_t stream)
{
  const float* x  = (const float*)d_in[0];
  const float* gW = (const float*)d_in[1];
  const float* gb = (const float*)d_in[2];
  const float* W1 = (const float*)d_in[3];
  const float* b1 = (const float*)d_in[4];
  const float* W2 = (const float*)d_in[5];
  const float* b2 = (const float*)d_in[6];
  float* out = (float*)d_out;

  int T = in_sizes[0] / D;                       // 16384
  int maxTiles = (T + TM - 1) / TM + NEXP;       // 520

  char* ws = (char*)d_ws;
  int*   topi        = (int*)ws;    ws += sizeof(int)   * 2 * (size_t)T;
  float* topw        = (float*)ws;  ws += sizeof(float) * 2 * (size_t)T;
  int*   counts      = (int*)ws;    ws += sizeof(int)   * 2 * NEXP;
  int*   cursor      = (int*)ws;    ws += sizeof(int)   * 2 * NEXP;
  int*   offsets     = (int*)ws;    ws += sizeof(int)   * 2 * NEXP;
  int*   lists       = (int*)ws;    ws += sizeof(int)   * 2 * (size_t)T;
  int*   tileCount   = (int*)ws;    ws += sizeof(int)   * 2;
  int*   tile_expert = (int*)ws;    ws += sizeof(int)   * 2 * (size_t)maxTiles;
  int*   tile_start  = (int*)ws;    ws += sizeof(int)   * 2 * (size_t)maxTiles;
  int*   tile_nvalid = (int*)ws;    ws += sizeof(int)   * 2 * (size_t)maxTiles;

  init_counts<<<1, 64, 0, stream>>>(counts);
  gate_kernel<<<(T + 7) / 8, 256, 0, stream>>>(x, gW, gb, topi, topw, counts, T);
  scan_kernel<<<1, 1, 0, stream>>>(counts, offsets, tileCount,
                                   tile_expert, tile_start, tile_nvalid, maxTiles);
  scatter_kernel<<<(T + 255) / 256, 256, 0, stream>>>(topi, offsets, cursor, lists, T);

  size_t ldsBytes = (size_t)LDS_FLOATS * sizeof(float);   // 208896 B (< 320 KB/WG)
  moe_gemm_kernel<<<maxTiles, 256, ldsBytes, stream>>>(x, W1, b1, W2, b2, out,
      tileCount, tile_expert, tile_start, tile_nvalid, lists, topw, T, maxTiles, 0);
  moe_gemm_kernel<<<maxTiles, 256, ldsBytes, stream>>>(x, W1, b1, W2, b2, out,
      tileCount, tile_expert, tile_start, tile_nvalid, lists, topw, T, maxTiles, 1);
}